// STTemporalSpatialModule_41034117546355
// MI455X (gfx1250) — compile-verified
//
#include <hip/hip_runtime.h>
#include <hip/hip_bf16.h>
#include <math.h>

// ---------------------------------------------------------------------------
// CDNA5 WMMA f32 16x16x4 fragment layouts (wave32):
//   A: 16x4 (MxK)  -> 2 VGPRs/lane : lanes 0-15 M=lane K={k,k+1}; lanes 16-31 K={k+2,k+3}
//   B: 4x16 (KxN)  -> 2 VGPRs/lane : lanes 0-15 N=lane K={k,k+1}; lanes 16-31 K={k+2,k+3}
//   C/D: 16x16     -> 8 VGPRs/lane : VGPR r, lane l: M = r + 8*(l>>4), N = l&15
// ---------------------------------------------------------------------------
typedef __attribute__((ext_vector_type(2))) float v2f;
typedef __attribute__((ext_vector_type(8))) float v8f;

#define BM 128   // block tile rows  (8 waves x 16)
#define BN 64    // block tile cols  (4 wmma tiles per wave)
#define BK 32    // k-chunk staged through LDS
#define SBP 80   // LDS pair-row pitch (v2f units); 80 = 16 (mod 32) -> the two
                 // lane-halves (rows kp, kp+1) land on disjoint bank halves

__device__ __forceinline__ v8f wmma4(v2f a, v2f b, v8f c) {
    return __builtin_amdgcn_wmma_f32_16x16x4_f32(false, a, false, b, (short)0, c, false, false);
}

// ---------------------------------------------------------------------------
// Generic fp32 GEMM via V_WMMA_F32_16X16X4_F32.
//   C[M,N] = A[M,K] (row stride lda) @ B[K,N] (+ bias[N])
//   transT > 0 : rows are (b, t), t in [0, transT); store C[b][n][t]
// All call sites: M % 128 == 0, K % 4 == 0, N % 16 == 0; ragged N/K vs the
// 64/32 tiles handled by zero-padded slow path (EXEC stays full for WMMA).
// ---------------------------------------------------------------------------
__global__ __launch_bounds__(256) void gemm_f32_wmma(
    const float* __restrict__ A, int lda,
    const float* __restrict__ B,
    float* __restrict__ C,
    const float* __restrict__ bias,
    int M, int N, int K, int transT)
{
    __shared__ v2f sB[(BK / 2) * SBP];   // K-pair-major: sB[kp*SBP+n] = {B[2kp][n], B[2kp+1][n]}

    const int tid   = threadIdx.x;
    const int wave  = tid >> 5;
    const int lane  = tid & 31;
    const int khalf = lane >> 4;   // 0 or 1
    const int l16   = lane & 15;
    const int n0    = blockIdx.x * BN;
    const int m0    = blockIdx.y * BM + wave * 16;

    v8f acc0 = {}; v8f acc1 = {}; v8f acc2 = {}; v8f acc3 = {};

    const float* Arow = A + (size_t)(m0 + l16) * lda;

    // Cooperative-load ownership (fast path): thread -> (pair-row, 4 columns).
    const int kp_ld = tid >> 4;          // 0..15
    const int nn_ld = (tid & 15) << 2;   // 0,4,...,60

    for (int k0 = 0; k0 < K; k0 += BK) {
        const int kvalid = K - k0;
        const bool full  = (kvalid >= BK) && (n0 + BN <= N);

        // Prefetch next B chunk into the cache hierarchy (global_prefetch_b8).
        if (k0 + BK < K) {
            const float* p = &B[(size_t)(k0 + BK + (tid >> 3)) * N + n0 + ((tid & 7) << 3)];
            __builtin_prefetch(p, 0, 1);
        }

        if (full) {
            // 2x global_load_b128 (adjacent K rows) -> 4x ds_store_b64 K-pairs.
            const float4 vlo = *(const float4*)&B[(size_t)(k0 + 2 * kp_ld    ) * N + n0 + nn_ld];
            const float4 vhi = *(const float4*)&B[(size_t)(k0 + 2 * kp_ld + 1) * N + n0 + nn_ld];
            v2f* dst = &sB[kp_ld * SBP + nn_ld];
            v2f p;
            p.x = vlo.x; p.y = vhi.x; dst[0] = p;
            p.x = vlo.y; p.y = vhi.y; dst[1] = p;
            p.x = vlo.z; p.y = vhi.z; dst[2] = p;
            p.x = vlo.w; p.y = vhi.w; dst[3] = p;
        } else {
            for (int i = tid; i < BK * BN; i += 256) {
                const int kk = i >> 6;
                const int nn = i & 63;
                float v = 0.f;
                if (kk < kvalid && (n0 + nn) < N)
                    v = B[(size_t)(k0 + kk) * N + (n0 + nn)];
                ((float*)&sB[(kk >> 1) * SBP + nn])[kk & 1] = v;
            }
        }
        __syncthreads();

        if (kvalid >= BK) {
#pragma unroll
            for (int kk = 0; kk < BK; kk += 4) {
                const v2f a = *(const v2f*)(Arow + k0 + kk + khalf * 2);   // b64
                const v2f* sBr = &sB[((kk >> 1) + khalf) * SBP];
                const v2f b0 = sBr[l16];
                const v2f b1 = sBr[l16 + 16];
                const v2f b2 = sBr[l16 + 32];
                const v2f b3 = sBr[l16 + 48];
                acc0 = wmma4(a, b0, acc0);
                acc1 = wmma4(a, b1, acc1);
                acc2 = wmma4(a, b2, acc2);
                acc3 = wmma4(a, b3, acc3);
            }
        } else {
#pragma unroll
            for (int kk = 0; kk < BK; kk += 4) {
                const int kb = k0 + kk + khalf * 2;
                v2f a; a.x = 0.f; a.y = 0.f;
                if (kb < K) a = *(const v2f*)(Arow + kb);
                const v2f* sBr = &sB[((kk >> 1) + khalf) * SBP];
                const v2f b0 = sBr[l16];
                const v2f b1 = sBr[l16 + 16];
                const v2f b2 = sBr[l16 + 32];
                const v2f b3 = sBr[l16 + 48];
                acc0 = wmma4(a, b0, acc0);
                acc1 = wmma4(a, b1, acc1);
                acc2 = wmma4(a, b2, acc2);
                acc3 = wmma4(a, b3, acc3);
            }
        }
        __syncthreads();
    }

    // Epilogue. Tiles never cross a batch boundary (transT % 16 == 0), so the
    // batch split is computed once; transposed rows become contiguous b128s.
    int bidx = 0, tt0 = 0;
    if (transT > 0) { bidx = m0 / transT; tt0 = m0 - bidx * transT; }

#pragma unroll
    for (int j = 0; j < 4; ++j) {
        const int col = n0 + l16 + 16 * j;
        if (col >= N) continue;
        const v8f acc = (j == 0) ? acc0 : (j == 1) ? acc1 : (j == 2) ? acc2 : acc3;
        const float bv = bias ? bias[col] : 0.f;
        if (transT > 0) {
            float* Cb = C + ((size_t)bidx * N + col) * transT + tt0 + 8 * khalf;
            float4 lo, hi;
            lo.x = acc[0] + bv; lo.y = acc[1] + bv; lo.z = acc[2] + bv; lo.w = acc[3] + bv;
            hi.x = acc[4] + bv; hi.y = acc[5] + bv; hi.z = acc[6] + bv; hi.w = acc[7] + bv;
            *(float4*)&Cb[0] = lo;          // global_store_b128
            *(float4*)&Cb[4] = hi;
        } else {
            float* Cb = C + (size_t)(m0 + 8 * khalf) * N + col;
#pragma unroll
            for (int rr = 0; rr < 8; ++rr) Cb[(size_t)rr * N] = acc[rr] + bv;
        }
    }
}

// ---------------------------------------------------------------------------
// Depthwise causal conv (D_CONV=4) over the sequence dim + SiLU.
// xz layout (b, T, 2*di); xh = xz[..., :di].  xc layout (b, T, di).
// ---------------------------------------------------------------------------
__global__ __launch_bounds__(256) void conv_silu_kernel(
    const float* __restrict__ xz, const float* __restrict__ w,
    const float* __restrict__ bias, float* __restrict__ xc,
    int T, int di)
{
    const size_t idx = (size_t)blockIdx.x * blockDim.x + threadIdx.x;
    const int    d   = (int)(idx % di);
    const size_t bt  = idx / di;
    const int    t   = (int)(bt % T);
    const size_t b   = bt / T;

    const int stride2 = 2 * di;
    const float* base = xz + (b * (size_t)T) * stride2 + d;
    const float* wd   = w + (size_t)d * 4;

    float acc = bias[d];
#pragma unroll
    for (int j = 0; j < 4; ++j) {
        const int ts = t - 3 + j;
        if (ts >= 0) acc += wd[j] * base[(size_t)ts * stride2];
    }
    xc[idx] = acc / (1.f + __expf(-acc));   // silu
}

// ---------------------------------------------------------------------------
// Selective scan, lane-per-state-element: 16 lanes = one channel's n-states,
// 2 channels per wave32 -> 16x the wave parallelism of thread-per-channel.
// Recurrence per lane: h = exp(dt*A_n)*h + (dt*x)*B_n   (1 exp + 2 FMA)
// Output contraction sum_n h*C_n via 4-step shfl_xor butterfly (off the
// recurrence critical path). Fuses softplus(dt), +x*D, *silu(z).
//   dtpre (b,T,di), xc (b,T,di), dbc (b,T,width) cols [r,r+16)=B [r+16,r+32)=C
//   xz (b,T,2di): z = xz[..., di:].  ya (b,T,di): gated output for out-GEMM.
// ---------------------------------------------------------------------------
__global__ __launch_bounds__(256) void scan_kernel(
    const float* __restrict__ dtpre, const float* __restrict__ xc,
    const float* __restrict__ dbc,   const float* __restrict__ xz,
    const float* __restrict__ A_log, const float* __restrict__ Dp,
    float* __restrict__ ya, int T, int di, int r, int width)
{
    const int n     = threadIdx.x & 15;        // state index
    const int group = threadIdx.x >> 4;        // 16 channel-groups per block
    const int d     = blockIdx.x * 16 + group; // channel
    const int b     = blockIdx.y;

    const float An = -__expf(A_log[(size_t)d * 16 + n]);
    const float Dd = Dp[d];
    float h = 0.f;

    const size_t row0 = (size_t)b * T;
    for (int t = 0; t < T; ++t) {
        const size_t base = row0 + t;
        const float dtp = dtpre[base * di + d];                        // broadcast
        const float dt  = (dtp > 20.f) ? dtp : log1pf(__expf(dtp));    // softplus
        const float x   = xc[base * di + d];                           // broadcast
        const float Bn  = dbc[base * width + r + n];
        const float Cn  = dbc[base * width + r + 16 + n];

        h = __expf(dt * An) * h + (dt * x) * Bn;

        float y = h * Cn;
        y += __shfl_xor(y, 1, 32);
        y += __shfl_xor(y, 2, 32);
        y += __shfl_xor(y, 4, 32);
        y += __shfl_xor(y, 8, 32);

        if (n == 0) {
            const float z  = xz[base * (size_t)(2 * di) + di + d];
            const float sz = z / (1.f + __expf(-z));
            ya[base * di + d] = (y + x * Dd) * sz;
        }
    }
}

// ---------------------------------------------------------------------------
static inline void run_gemm(const float* A, int lda, const float* B, float* C,
                            const float* bias, int M, int N, int K, int transT,
                            hipStream_t s)
{
    dim3 grid((N + BN - 1) / BN, M / BM);
    gemm_f32_wmma<<<grid, 256, 0, s>>>(A, lda, B, C, bias, M, N, K, transT);
}

extern "C" void kernel_launch(void* const* d_in, const int* in_sizes, int n_in,
                              void* d_out, int out_size, void* d_ws, size_t ws_size,
                              hipStream_t stream)
{
    (void)in_sizes; (void)n_in; (void)out_size; (void)ws_size;

    const float* x      = (const float*)d_in[0];    // (32, 256, 512)
    // stage 1 ("s_"): d_model=512, di=1024, r=32, T=256
    const float* s_in_w    = (const float*)d_in[1];   // (512, 2048)
    const float* s_conv_w  = (const float*)d_in[2];   // (1024, 1, 4)
    const float* s_conv_b  = (const float*)d_in[3];   // (1024,)
    const float* s_xproj_w = (const float*)d_in[4];   // (1024, 64)
    const float* s_dt_w    = (const float*)d_in[5];   // (32, 1024)
    const float* s_dt_b    = (const float*)d_in[6];   // (1024,)
    const float* s_A_log   = (const float*)d_in[7];   // (1024, 16)
    const float* s_D       = (const float*)d_in[8];   // (1024,)
    const float* s_out_w   = (const float*)d_in[9];   // (1024, 512)
    // stage 2 ("t_"): d_model=256, di=512, r=16, T=512
    const float* t_in_w    = (const float*)d_in[10];  // (256, 1024)
    const float* t_conv_w  = (const float*)d_in[11];  // (512, 1, 4)
    const float* t_conv_b  = (const float*)d_in[12];  // (512,)
    const float* t_xproj_w = (const float*)d_in[13];  // (512, 48)
    const float* t_dt_w    = (const float*)d_in[14];  // (16, 512)
    const float* t_dt_b    = (const float*)d_in[15];  // (512,)
    const float* t_A_log   = (const float*)d_in[16];  // (512, 16)
    const float* t_D       = (const float*)d_in[17];  // (512,)
    const float* t_out_w   = (const float*)d_in[18];  // (512, 256)

    float* out = (float*)d_out;                       // (32, 256, 512)

    // Workspace layout (floats); buffers reused between the two stages.
    float* ws = (float*)d_ws;
    size_t off = 0;
    float* xz    = ws + off; off += 16777216;  // max(32*256*2048, 32*512*1024)
    float* xcb   = ws + off; off += 8388608;   // max(32*256*1024, 32*512*512)
    float* dbc   = ws + off; off += 786432;    // max(32*256*64,  32*512*48)
    float* dtpre = ws + off; off += 8388608;
    float* yab   = ws + off; off += 8388608;
    float* o1t   = ws + off; off += 4194304;   // (32, 512, 256)

    // ---------------- Stage 1 (spatial) ----------------
    {
        const int B = 32, T = 256, dm = 512, di = 1024, r = 32, w = 64;
        const int M = B * T;                               // 8192
        run_gemm(x, dm, s_in_w, xz, nullptr, M, 2 * di, dm, 0, stream);
        conv_silu_kernel<<<(size_t)M * di / 256, 256, 0, stream>>>(xz, s_conv_w, s_conv_b, xcb, T, di);
        run_gemm(xcb, di, s_xproj_w, dbc, nullptr, M, w, di, 0, stream);
        run_gemm(dbc, w, s_dt_w, dtpre, s_dt_b, M, di, r, 0, stream);
        scan_kernel<<<dim3(di / 16, B), 256, 0, stream>>>(dtpre, xcb, dbc, xz, s_A_log, s_D, yab, T, di, r, w);
        run_gemm(yab, di, s_out_w, o1t, nullptr, M, dm, di, /*transT=*/T, stream);   // -> (B, 512, 256)
    }

    // ---------------- Stage 2 (temporal) ----------------
    {
        const int B = 32, T = 512, dm = 256, di = 512, r = 16, w = 48;
        const int M = B * T;                               // 16384
        run_gemm(o1t, dm, t_in_w, xz, nullptr, M, 2 * di, dm, 0, stream);
        conv_silu_kernel<<<(size_t)M * di / 256, 256, 0, stream>>>(xz, t_conv_w, t_conv_b, xcb, T, di);
        run_gemm(xcb, di, t_xproj_w, dbc, nullptr, M, w, di, 0, stream);
        run_gemm(dbc, w, t_dt_w, dtpre, t_dt_b, M, di, r, 0, stream);
        scan_kernel<<<dim3(di / 16, B), 256, 0, stream>>>(dtpre, xcb, dbc, xz, t_A_log, t_D, yab, T, di, r, w);
        run_gemm(yab, di, t_out_w, out, nullptr, M, dm, di, /*transT=*/T, stream);   // -> (32, 256, 512)
    }
}